// R_Sinkhorn_Optimal_Transport_87067577024970
// MI455X (gfx1250) — compile-verified
//
#include <hip/hip_runtime.h>
#include <math.h>

// Sinkhorn log-domain normalization, MI455X (gfx1250, wave32).
// x:[4,2048,2048] f32 (67 MB, L2-resident on 192MB L2). 10 iterations of
// row/col logsumexp + final exp pass. log_K = -x/EPS recomputed on the fly.

typedef __attribute__((ext_vector_type(2))) float v2f;
typedef __attribute__((ext_vector_type(8))) float v8f;

#define SK_B     4
#define SK_N     2048
#define SK_M     2048
#define SK_ITERS 10
#define INV_EPS  20.0f                    // 1/0.05
#define LOG_MU   (-7.6246189861593985f)   // -log(2048)
#define LOG_NU   (-7.6246189861593985f)
#define SEG      256                      // rows per column-segment
#define NSEG     (SK_N / SEG)             // 8

// ---------------------------------------------------------------------------
// Exact f32 sum of the 32 per-lane partials of a wave, via a chain of
// V_WMMA_F32_16X16X4_F32 (D = A*B + C) with a ones matrix.
//   WMMA1: A[lane-layout] holds s in K0 (lanes 0-15) / K2 (lanes 16-31),
//          zeros elsewhere; B = ones  =>  D[m,*] = s_m + s_{m+16}.
//   WMMA2..5: A = ones, B = consecutive D VGPR pairs, C chained
//          => every D VGPR half (a constant row) is contracted exactly once,
//          so acc[*] = sum of all 32 lane partials, broadcast everywhere.
// EXEC must be all-ones: callers run with full blocks and no divergence.
// ---------------------------------------------------------------------------
__device__ __forceinline__ float wave_sum32_wmma(float s) {
  v2f a;   a[0] = s;    a[1] = 0.0f;
  v2f one; one[0] = 1.0f; one[1] = 1.0f;
  v8f zero = {};
  v8f d = __builtin_amdgcn_wmma_f32_16x16x4_f32(
      false, a, false, one, (short)0, zero, false, false);
  v8f acc = {};
#pragma unroll
  for (int r = 0; r < 8; r += 2) {
    v2f bp; bp[0] = d[r]; bp[1] = d[r + 1];
    acc = __builtin_amdgcn_wmma_f32_16x16x4_f32(
        false, one, false, bp, (short)0, acc, false, false);
  }
  return acc[0];
}

__device__ __forceinline__ float wave_max32(float m) {
#pragma unroll
  for (int k = 16; k >= 1; k >>= 1) m = fmaxf(m, __shfl_xor(m, k, 32));
  return m;
}

// ---------------------------------------------------------------------------
__global__ __launch_bounds__(256) void sinkhorn_init_v(float* __restrict__ v) {
  int i = blockIdx.x * blockDim.x + threadIdx.x;
  if (i < SK_B * SK_M) v[i] = 0.0f;
}

// u[b,i] = LOG_MU - logsumexp_j( -x[b,i,j]/EPS + v[b,j] )
// One wave per row (2048 f32 = 64 regs/lane), 8 waves/block, coalesced b128.
__global__ __launch_bounds__(256) void sinkhorn_row_update(
    const float* __restrict__ x, const float* __restrict__ v,
    float* __restrict__ u) {
  const int wave = threadIdx.x >> 5;
  const int lane = threadIdx.x & 31;
  const int row  = blockIdx.x * 8 + wave;     // global row in [0, B*N)
  const int b    = row >> 11;                 // row / 2048
  const float* __restrict__ xr = x + (size_t)row * SK_M;
  const float* __restrict__ vb = v + b * SK_M;

  float4 t[16];
  float m = -INFINITY;
#pragma unroll
  for (int i = 0; i < 16; ++i) {
    const int c = i * 128 + lane * 4;
    const float4 xv = *(const float4*)(xr + c);
    const float4 vv = *(const float4*)(vb + c);
    float4 tt;
    tt.x = fmaf(xv.x, -INV_EPS, vv.x);
    tt.y = fmaf(xv.y, -INV_EPS, vv.y);
    tt.z = fmaf(xv.z, -INV_EPS, vv.z);
    tt.w = fmaf(xv.w, -INV_EPS, vv.w);
    t[i] = tt;
    m = fmaxf(m, fmaxf(fmaxf(tt.x, tt.y), fmaxf(tt.z, tt.w)));
  }
  m = wave_max32(m);

  float s = 0.0f;
#pragma unroll
  for (int i = 0; i < 16; ++i) {
    s += __expf(t[i].x - m);
    s += __expf(t[i].y - m);
    s += __expf(t[i].z - m);
    s += __expf(t[i].w - m);
  }
  const float S = wave_sum32_wmma(s);
  if (lane == 0) u[row] = LOG_MU - (m + __logf(S));
}

// Per-(segment, column) partial logsumexp of ( -x[b,i,j]/EPS + u[b,i] ) over i.
// Thread-per-column => coalesced row reads; pass 2 re-reads hot L0/L2 data.
__global__ __launch_bounds__(256) void sinkhorn_col_partial(
    const float* __restrict__ x, const float* __restrict__ u,
    float* __restrict__ wm, float* __restrict__ ws) {
  const int ctiles = SK_M / 256;
  const int blk   = blockIdx.x;
  const int ctile = blk % ctiles;
  const int seg   = (blk / ctiles) % NSEG;
  const int b     = blk / (ctiles * NSEG);
  const int c     = ctile * 256 + threadIdx.x;
  const float* __restrict__ xb = x + (size_t)b * SK_N * SK_M;
  const float* __restrict__ ub = u + b * SK_N;
  const int r0 = seg * SEG;

  float m = -INFINITY;
#pragma unroll 8
  for (int r = 0; r < SEG; ++r) {
    const float t = fmaf(xb[(size_t)(r0 + r) * SK_M + c], -INV_EPS, ub[r0 + r]);
    m = fmaxf(m, t);
  }
  float s = 0.0f;
#pragma unroll 8
  for (int r = 0; r < SEG; ++r) {
    const float t = fmaf(xb[(size_t)(r0 + r) * SK_M + c], -INV_EPS, ub[r0 + r]);
    s += __expf(t - m);
  }
  const int o = (b * NSEG + seg) * SK_M + c;
  wm[o] = m;
  ws[o] = s;
}

// v[b,j] = LOG_NU - merge of NSEG (m,s) logsumexp partials.
__global__ __launch_bounds__(256) void sinkhorn_col_combine(
    const float* __restrict__ wm, const float* __restrict__ ws,
    float* __restrict__ v) {
  const int i = blockIdx.x * blockDim.x + threadIdx.x;  // [0, B*M)
  if (i >= SK_B * SK_M) return;
  const int b = i >> 11;
  const int c = i & (SK_M - 1);
  float m = -INFINITY;
#pragma unroll
  for (int s = 0; s < NSEG; ++s)
    m = fmaxf(m, wm[(b * NSEG + s) * SK_M + c]);
  float acc = 0.0f;
#pragma unroll
  for (int s = 0; s < NSEG; ++s) {
    const int o = (b * NSEG + s) * SK_M + c;
    acc += ws[o] * __expf(wm[o] - m);
  }
  v[i] = LOG_NU - (m + __logf(acc));
}

// P = exp( -x/EPS + u[:,:,None] + v[:,None,:] ), float4 per thread.
__global__ __launch_bounds__(256) void sinkhorn_final(
    const float* __restrict__ x, const float* __restrict__ u,
    const float* __restrict__ v, float* __restrict__ p) {
  const size_t e = ((size_t)blockIdx.x * blockDim.x + threadIdx.x) * 4;
  const int row = (int)(e >> 11);      // global row (M=2048)
  const int b   = row >> 11;
  const int c   = (int)(e & (SK_M - 1));
  const float4 xv = *(const float4*)(x + e);
  const float4 vv = *(const float4*)(v + b * SK_M + c);
  const float  uu = u[row];
  float4 o;
  o.x = __expf(fmaf(xv.x, -INV_EPS, uu + vv.x));
  o.y = __expf(fmaf(xv.y, -INV_EPS, uu + vv.y));
  o.z = __expf(fmaf(xv.z, -INV_EPS, uu + vv.z));
  o.w = __expf(fmaf(xv.w, -INV_EPS, uu + vv.w));
  *(float4*)(p + e) = o;
}

// ---------------------------------------------------------------------------
extern "C" void kernel_launch(void* const* d_in, const int* in_sizes, int n_in,
                              void* d_out, int out_size, void* d_ws, size_t ws_size,
                              hipStream_t stream) {
  const float* x = (const float*)d_in[0];
  float* P = (float*)d_out;

  // workspace layout (f32): ~576 KB total
  float* wm   = (float*)d_ws;                    // [B*NSEG*M]
  float* wsum = wm   + SK_B * NSEG * SK_M;       // [B*NSEG*M]
  float* u    = wsum + SK_B * NSEG * SK_M;       // [B*N]
  float* v    = u    + SK_B * SK_N;              // [B*M]

  const int rowBlocks = SK_B * SK_N / 8;                     // 1024
  const int colBlocks = SK_B * NSEG * (SK_M / 256);          // 256
  const int vecBlocks = (SK_B * SK_M + 255) / 256;           // 32
  const int finBlocks = (int)(((size_t)SK_B * SK_N * SK_M / 4) / 256); // 16384

  sinkhorn_init_v<<<vecBlocks, 256, 0, stream>>>(v);
  for (int it = 0; it < SK_ITERS; ++it) {
    sinkhorn_row_update<<<rowBlocks, 256, 0, stream>>>(x, v, u);
    sinkhorn_col_partial<<<colBlocks, 256, 0, stream>>>(x, u, wm, wsum);
    sinkhorn_col_combine<<<vecBlocks, 256, 0, stream>>>(wm, wsum, v);
  }
  sinkhorn_final<<<finBlocks, 256, 0, stream>>>(x, u, v, P);
}